// GatJumpStatPool_69346541961799
// MI455X (gfx1250) — compile-verified
//
#include <hip/hip_runtime.h>
#include <math.h>

#define DIM 128
#define NFEAT 9
#define NSLOPE 0.2f
#define BLK 256

typedef __attribute__((ext_vector_type(2))) float v2f;
typedef __attribute__((ext_vector_type(8))) float v8f;

__device__ inline void atomAddF(float* p, float v) {
  __hip_atomic_fetch_add(p, v, __ATOMIC_RELAXED, __HIP_MEMORY_SCOPE_AGENT);
}
__device__ inline void atomMaxF(float* p, float v) {
  __hip_atomic_fetch_max(p, v, __ATOMIC_RELAXED, __HIP_MEMORY_SCOPE_AGENT);
}
__device__ inline void atomMinF(float* p, float v) {
  __hip_atomic_fetch_min(p, v, __ATOMIC_RELAXED, __HIP_MEMORY_SCOPE_AGENT);
}
__device__ inline float lrelu(float v) { return v > 0.f ? v : NSLOPE * v; }

// ---------------------------------------------------------------------------
// Node GEMM: C[M,128] = A[M,128] @ B[128,128]. One wave per 16-row strip;
// the A fragment (2 VGPRs) is reused across all 8 column tiles -> 8
// independent-accumulator WMMAs per k-step (no RAW hazards, full XDL pipe).
// ---------------------------------------------------------------------------
__global__ __launch_bounds__(BLK)
void gemm_f32_wmma_strip(const float* __restrict__ A, const float* __restrict__ B,
                         float* __restrict__ C, int M, int K) {
  int wave = (int)((blockIdx.x * blockDim.x + threadIdx.x) >> 5);
  int lane = threadIdx.x & 31;
  int tM = wave << 4;
  if (tM >= M) return;
  int half = lane >> 4;  // 0: lanes 0-15, 1: lanes 16-31
  int l16  = lane & 15;
  // A 16x4 f32 layout: v0 = K=k0 (lo) / k0+2 (hi); v1 = k0+1 / k0+3
  const float* arow = A + (size_t)(tM + l16) * K + (half << 1);
  // B 4x16 layout mirrored: v0 = row k0 / k0+2 ; v1 = row k0+1 / k0+3
  const float* brow = B + (size_t)(half << 1) * DIM + l16;
  v8f acc[8] = {};
  for (int k0 = 0; k0 < K; k0 += 4) {
    v2f a; a.x = arow[0]; a.y = arow[1];
#pragma unroll
    for (int t = 0; t < 8; ++t) {
      v2f b; b.x = brow[t * 16]; b.y = brow[t * 16 + DIM];
      acc[t] = __builtin_amdgcn_wmma_f32_16x16x4_f32(false, a, false, b,
                                                     (short)0, acc[t], false, false);
    }
    arow += 4;
    brow += (size_t)4 * DIM;
  }
  // D layout: VGPR v -> row tM + v + 8*half, col = tile*16 + l16
  float* crow = C + (size_t)(tM + (half << 3)) * DIM + l16;
#pragma unroll
  for (int t = 0; t < 8; ++t)
#pragma unroll
    for (int v = 0; v < 8; ++v) crow[(size_t)v * DIM + t * 16] = acc[t][v];
}

// ---------------------------------------------------------------------------
// Generic tile GEMM (head): C[M,N] = A[M,K] @ B[K,N] + bias, 16x16 per wave.
// ---------------------------------------------------------------------------
__global__ __launch_bounds__(BLK)
void gemm_f32_wmma_tile(const float* __restrict__ A, const float* __restrict__ B,
                        const float* __restrict__ bias, float* __restrict__ C,
                        int M, int N, int K) {
  int wave = (int)((blockIdx.x * blockDim.x + threadIdx.x) >> 5);
  int lane = threadIdx.x & 31;
  int tilesN = N >> 4;
  int tM = (wave / tilesN) << 4;
  int tN = (wave % tilesN) << 4;
  if (tM >= M) return;
  int half = lane >> 4;
  int l16  = lane & 15;
  const float* arow = A + (size_t)(tM + l16) * K + (half << 1);
  const float* bcol = B + (size_t)(half << 1) * N + tN + l16;
  v8f acc = {};
  for (int k0 = 0; k0 < K; k0 += 4) {
    v2f a; a.x = arow[0]; a.y = arow[1];
    v2f b; b.x = bcol[0]; b.y = bcol[N];
    acc = __builtin_amdgcn_wmma_f32_16x16x4_f32(false, a, false, b,
                                                (short)0, acc, false, false);
    arow += 4;
    bcol += (size_t)4 * N;
  }
  float bv = bias ? bias[tN + l16] : 0.f;
  float* crow = C + (size_t)(tM + (half << 3)) * N + tN + l16;
#pragma unroll
  for (int v = 0; v < 8; ++v) crow[(size_t)v * N] = acc[v] + bv;
}

// ---------------------------------------------------------------------------
// AtomEncoder: h[n,k] = sum_f emb[f, x[n,f], k]
// ---------------------------------------------------------------------------
__global__ __launch_bounds__(BLK)
void embed_kernel(const int* __restrict__ x, const float* __restrict__ emb,
                  float* __restrict__ h, int N, int V) {
  int i = blockIdx.x * blockDim.x + threadIdx.x;
  if (i >= N * DIM) return;
  int n = i >> 7, k = i & (DIM - 1);
  float s = 0.f;
#pragma unroll
  for (int f = 0; f < NFEAT; ++f) {
    int idx = x[n * NFEAT + f];
    s += emb[((size_t)f * V + idx) * DIM + k];
  }
  h[i] = s;
}

// as[n] = hW[n,:]·att_src ; ad[n] = hW[n,:]·att_dst
__global__ __launch_bounds__(BLK)
void attn_proj_kernel(const float* __restrict__ hW, const float* __restrict__ a_s,
                      const float* __restrict__ a_d, float* __restrict__ as,
                      float* __restrict__ ad, int N) {
  int n = blockIdx.x * blockDim.x + threadIdx.x;
  if (n >= N) return;
  const float4* hp = (const float4*)(hW + (size_t)n * DIM);
  const float4* sp = (const float4*)a_s;
  const float4* dp = (const float4*)a_d;
  float ss = 0.f, dd = 0.f;
#pragma unroll
  for (int i = 0; i < DIM / 4; ++i) {
    float4 h = hp[i], s = sp[i], d = dp[i];
    ss += h.x * s.x + h.y * s.y + h.z * s.z + h.w * s.w;
    dd += h.x * d.x + h.y * d.y + h.z * d.z + h.w * d.w;
  }
  as[n] = ss;
  ad[n] = dd;
}

__global__ __launch_bounds__(BLK)
void init_mz_kernel(float* __restrict__ m, float* __restrict__ z, int N) {
  int n = blockIdx.x * blockDim.x + threadIdx.x;
  if (n >= N) return;
  m[n] = -__builtin_inff();
  z[n] = 0.f;
}

// segment max of leaky_relu(as[src]+ad[dst]) over dst (self-loops appended)
__global__ __launch_bounds__(BLK)
void edge_max_kernel(const int* __restrict__ ei, int E, int N,
                     const float* __restrict__ as, const float* __restrict__ ad,
                     float* __restrict__ m) {
  int i = blockIdx.x * blockDim.x + threadIdx.x;
  if (i >= E + N) return;
  int s = i < E ? ei[i] : i - E;
  int d = i < E ? ei[E + i] : i - E;
  atomMaxF(&m[d], lrelu(as[s] + ad[d]));
}

// one wave per edge: p = exp(e - m[dst]); z[dst]+=p; acc[dst,:]+=p*hW[src,:]
__global__ __launch_bounds__(BLK)
void edge_accum_kernel(const int* __restrict__ ei, int E, int N,
                       const float* __restrict__ as, const float* __restrict__ ad,
                       const float* __restrict__ m, const float* __restrict__ hW,
                       float* __restrict__ z, float* __restrict__ acc) {
  int gid = blockIdx.x * blockDim.x + threadIdx.x;
  int e = gid >> 5;
  int lane = gid & 31;
  if (e >= E + N) return;
  int s = e < E ? ei[e] : e - E;
  int d = e < E ? ei[E + e] : e - E;
  float p = __expf(lrelu(as[s] + ad[d]) - m[d]);
  if (lane == 0) atomAddF(&z[d], p);
  float4 hv = ((const float4*)(hW + (size_t)s * DIM))[lane];  // 32 lanes x 4
  float* ap = acc + (size_t)d * DIM + (lane << 2);
  atomAddF(&ap[0], p * hv.x);
  atomAddF(&ap[1], p * hv.y);
  atomAddF(&ap[2], p * hv.z);
  atomAddF(&ap[3], p * hv.w);
}

// h[n,k] = relu(acc[n,k]/max(z[n],1e-16) + b[k])
__global__ __launch_bounds__(BLK)
void node_update_kernel(const float* __restrict__ acc, const float* __restrict__ z,
                        const float* __restrict__ b, float* __restrict__ h, int N) {
  int i = blockIdx.x * blockDim.x + threadIdx.x;
  if (i >= N * DIM) return;
  int n = i >> 7, k = i & (DIM - 1);
  float zz = z[n];
  zz = zz > 1e-16f ? zz : 1e-16f;
  float v = acc[i] / zz + b[k];
  h[i] = v > 0.f ? v : 0.f;
}

// pooled layout: [B, (L+1)*3*DIM]; per jk entry j: [min(DIM) | sum(DIM) | max(DIM)]
__global__ __launch_bounds__(BLK)
void init_pool_kernel(float* __restrict__ pooled, float* __restrict__ counts,
                      int B, int POOLW) {
  int i = blockIdx.x * blockDim.x + threadIdx.x;
  if (i < B) counts[i] = 0.f;
  if (i >= B * POOLW) return;
  int r = i % (3 * DIM);
  pooled[i] = (r < DIM) ? __builtin_inff() : (r < 2 * DIM ? 0.f : -__builtin_inff());
}

__global__ __launch_bounds__(BLK)
void count_kernel(const int* __restrict__ batch, float* __restrict__ counts, int N) {
  int n = blockIdx.x * blockDim.x + threadIdx.x;
  if (n >= N) return;
  atomAddF(&counts[batch[n]], 1.f);
}

__global__ __launch_bounds__(BLK)
void statpool_kernel(const float* __restrict__ h, const int* __restrict__ batch,
                     float* __restrict__ pooled, int N, int j, int POOLW) {
  int i = blockIdx.x * blockDim.x + threadIdx.x;
  if (i >= N * DIM) return;
  int n = i >> 7, k = i & (DIM - 1);
  int g = batch[n];
  float v = h[i];
  float* base = pooled + (size_t)g * POOLW + j * 3 * DIM;
  atomMinF(&base[k], v);
  atomAddF(&base[DIM + k], v);
  atomMaxF(&base[2 * DIM + k], v);
}

__global__ __launch_bounds__(BLK)
void finalize_pool_kernel(float* __restrict__ pooled, const float* __restrict__ counts,
                          int B, int NJ, int POOLW) {
  int i = blockIdx.x * blockDim.x + threadIdx.x;
  if (i >= B * NJ * DIM) return;
  int g = i / (NJ * DIM);
  int r = i % (NJ * DIM);
  int j = r / DIM, k = r % DIM;
  float c = counts[g];
  if (c < 1.f) c = 1.f;
  pooled[(size_t)g * POOLW + j * 3 * DIM + DIM + k] /= c;
}

static inline int cdiv(long long a, int b) { return (int)((a + b - 1) / b); }

extern "C" void kernel_launch(void* const* d_in, const int* in_sizes, int n_in,
                              void* d_out, int out_size, void* d_ws, size_t ws_size,
                              hipStream_t stream) {
  const int*   x     = (const int*)d_in[0];
  const int*   ei    = (const int*)d_in[1];
  const int*   batch = (const int*)d_in[2];
  const float* emb   = (const float*)d_in[4];
  const float* convW = (const float*)d_in[5];
  const float* attS  = (const float*)d_in[6];
  const float* attD  = (const float*)d_in[7];
  const float* convB = (const float*)d_in[8];
  const float* linW  = (const float*)d_in[9];
  const float* linb  = (const float*)d_in[10];
  float* out = (float*)d_out;

  const int N = in_sizes[0] / NFEAT;           // 50000
  const int E = in_sizes[1] / 2;               // 1600000
  const int L = in_sizes[5] / (DIM * DIM);     // 5
  const int V = in_sizes[4] / (NFEAT * DIM);   // 64
  const int OUT = in_sizes[10];                // 128
  const int POOLW = in_sizes[9] / OUT;         // 2304
  const int B = out_size / OUT;                // 128
  const int EE = E + N;                        // edges incl self loops

  float* ws = (float*)d_ws;
  float* h      = ws;                          // [N, DIM]
  float* hW     = h   + (size_t)N * DIM;       // [N, DIM]
  float* acc    = hW  + (size_t)N * DIM;       // [N, DIM]
  float* as     = acc + (size_t)N * DIM;       // [N]
  float* ad     = as + N;                      // [N]
  float* m      = ad + N;                      // [N]
  float* z      = m + N;                       // [N]
  float* pooled = z + N;                       // [B, POOLW]
  float* counts = pooled + (size_t)B * POOLW;  // [B]

  // pools + counts
  init_pool_kernel<<<cdiv((long long)B * POOLW, BLK), BLK, 0, stream>>>(pooled, counts, B, POOLW);
  count_kernel<<<cdiv(N, BLK), BLK, 0, stream>>>(batch, counts, N);

  // atom encoder
  embed_kernel<<<cdiv((long long)N * DIM, BLK), BLK, 0, stream>>>(x, emb, h, N, V);
  statpool_kernel<<<cdiv((long long)N * DIM, BLK), BLK, 0, stream>>>(h, batch, pooled, N, 0, POOLW);

  const int stripWaves = N / 16;               // M strips of 16 rows
  for (int l = 0; l < L; ++l) {
    gemm_f32_wmma_strip<<<cdiv((long long)stripWaves * 32, BLK), BLK, 0, stream>>>(
        h, convW + (size_t)l * DIM * DIM, hW, N, DIM);
    attn_proj_kernel<<<cdiv(N, BLK), BLK, 0, stream>>>(
        hW, attS + (size_t)l * DIM, attD + (size_t)l * DIM, as, ad, N);
    init_mz_kernel<<<cdiv(N, BLK), BLK, 0, stream>>>(m, z, N);
    hipMemsetAsync(acc, 0, (size_t)N * DIM * sizeof(float), stream);
    edge_max_kernel<<<cdiv(EE, BLK), BLK, 0, stream>>>(ei, E, N, as, ad, m);
    edge_accum_kernel<<<cdiv((long long)EE * 32, BLK), BLK, 0, stream>>>(
        ei, E, N, as, ad, m, hW, z, acc);
    node_update_kernel<<<cdiv((long long)N * DIM, BLK), BLK, 0, stream>>>(
        acc, z, convB + (size_t)l * DIM, h, N);
    statpool_kernel<<<cdiv((long long)N * DIM, BLK), BLK, 0, stream>>>(
        h, batch, pooled, N, l + 1, POOLW);
  }

  finalize_pool_kernel<<<cdiv((long long)B * (L + 1) * DIM, BLK), BLK, 0, stream>>>(
      pooled, counts, B, L + 1, POOLW);

  // head: out[B, OUT] = pooled[B, POOLW] @ linW[POOLW, OUT] + linb
  const int headWaves = (B / 16) * (OUT / 16);
  gemm_f32_wmma_tile<<<cdiv((long long)headWaves * 32, BLK), BLK, 0, stream>>>(
      pooled, linW, linb, out, B, OUT, POOLW);
}